// CubicSplinePSF_58222576665189
// MI455X (gfx1250) — compile-verified
//
#include <hip/hip_runtime.h>

typedef float v2f __attribute__((ext_vector_type(2)));
typedef float v8f __attribute__((ext_vector_type(8)));

#define ROISIZE 20
#define ZPLANES 8
#define SPL_D 64
#define SPL_H 40
#define SPL_W 40
#define NPTS   (ZPLANES * ROISIZE * ROISIZE)   /* 3200 points per emitter */
#define NTILES (NPTS / 16)                     /* 200 tiles of 16 points  */

__global__ __launch_bounds__(256) void psf_wmma_kernel(
    const float* __restrict__ pos,    // [N,3]
    const float* __restrict__ inten,  // [N,8]
    const float* __restrict__ bg,     // [N,8]
    const float* __restrict__ coefs,  // [64,40,40,64]
    float* __restrict__ out)          // [N,8,20,20]
{
    __shared__ float s_raw[NPTS];
    __shared__ float s_basis[64];
    __shared__ float s_scale[ZPLANES];
    __shared__ float s_off[ZPLANES];

    const int n    = blockIdx.x;
    const int tid  = threadIdx.x;
    const int lane = tid & 31;
    const int wave = tid >> 5;     // 0..7
    const int nl   = lane & 15;    // column / point within tile
    const int hl   = lane >> 4;    // half-wave selects K pair

    // ---- per-emitter fractional parts & integer cell bases ----
    const float p0 = pos[n * 3 + 0];  // indexes y (faithful to reference)
    const float p1 = pos[n * 3 + 1];  // indexes x
    const float p2 = pos[n * 3 + 2];  // indexes z
    const float tz = 28.0f - p2; const float fz = floorf(tz); const float dz = tz - fz;
    const float ty = 10.0f - p0; const float fy = floorf(ty); const float dy = ty - fy;
    const float tx = 10.0f - p1; const float fx = floorf(tx); const float dx = tx - fx;
    const int iz0 = (int)fz, iy0 = (int)fy, ix0 = (int)fx;

    // ---- 64-entry tricubic basis, one vector per emitter ----
    if (tid < 64) {
        const int kz = tid >> 4, ky = (tid >> 2) & 3, kx = tid & 3;
        float bz = 1.f; for (int i = 0; i < kz; ++i) bz *= dz;
        float by = 1.f; for (int i = 0; i < ky; ++i) by *= dy;
        float bx = 1.f; for (int i = 0; i < kx; ++i) bx *= dx;
        s_basis[tid] = bz * by * bx;
    }
    __syncthreads();

    // A operand: basis chunk replicated across all 16 rows.
    // Lane layout (16x4 f32 A): vgpr v, half hl -> k = 4c + v + 2*hl.
    v2f aop[16];
#pragma unroll
    for (int c = 0; c < 16; ++c) {
        aop[c].x = s_basis[4 * c + 2 * hl + 0];
        aop[c].y = s_basis[4 * c + 2 * hl + 1];
    }

    // ---- raw spline values: GEMV as chained 16x16x4 f32 WMMAs ----
    for (int t = wave; t < NTILES; t += 8) {   // 25 uniform iterations per wave
        const int p   = t * 16 + nl;
        const int z   = p / 400;
        const int rem = p - z * 400;
        const int y   = rem / 20;
        const int x   = rem - y * 20;
        int iz = iz0 + z; iz = iz < 0 ? 0 : (iz > SPL_D - 1 ? SPL_D - 1 : iz);
        int iy = iy0 + y; iy = iy < 0 ? 0 : (iy > SPL_H - 1 ? SPL_H - 1 : iy);
        int ix = ix0 + x; ix = ix < 0 ? 0 : (ix > SPL_W - 1 ? SPL_W - 1 : ix);
        const float* cbase =
            coefs + (((size_t)iz * SPL_H + (size_t)iy) * SPL_W + (size_t)ix) * 64;

        v8f acc = {0.f, 0.f, 0.f, 0.f, 0.f, 0.f, 0.f, 0.f};
#pragma unroll
        for (int c = 0; c < 16; ++c) {
            // B operand (4x16 K x N): lane nl = point, k = 4c + 2*hl + {0,1}
            const v2f b = *(const v2f*)(cbase + 4 * c + 2 * hl);
            acc = __builtin_amdgcn_wmma_f32_16x16x4_f32(
                /*neg_a=*/false, aop[c], /*neg_b=*/false, b,
                /*c_mod=*/(short)0, acc, /*reuse_a=*/false, /*reuse_b=*/false);
        }
        // Every D row is identical; low half-wave writes column nl once.
        if (hl == 0) s_raw[p] = acc[0];
    }
    __syncthreads();

    // ---- per-plane sums: wave w reduces plane w ----
    float psum = 0.f;
    for (int i = lane; i < ROISIZE * ROISIZE; i += 32)
        psum += s_raw[wave * 400 + i];
#pragma unroll
    for (int off = 16; off >= 1; off >>= 1)
        psum += __shfl_xor(psum, off, 32);
    if (lane == 0) {
        s_scale[wave] = inten[n * ZPLANES + wave] / psum;
        s_off[wave]   = bg[n * ZPLANES + wave];
    }
    __syncthreads();

    // ---- normalize + intensity/background, coalesced store ----
    float* obase = out + (size_t)n * NPTS;
    for (int i = tid; i < NPTS; i += 256) {
        const int z = i / 400;
        obase[i] = s_raw[i] * s_scale[z] + s_off[z];
    }
}

extern "C" void kernel_launch(void* const* d_in, const int* in_sizes, int n_in,
                              void* d_out, int out_size, void* d_ws, size_t ws_size,
                              hipStream_t stream) {
    const float* pos   = (const float*)d_in[0];
    const float* inten = (const float*)d_in[1];
    const float* bg    = (const float*)d_in[2];
    const float* coefs = (const float*)d_in[3];
    const int N = in_sizes[0] / 3;   // 512 emitters
    psf_wmma_kernel<<<N, 256, 0, stream>>>(pos, inten, bg, coefs, (float*)d_out);
}